// PPoolLoss_89335319757026
// MI455X (gfx1250) — compile-verified
//
#include <hip/hip_runtime.h>
#include <math.h>

// Problem constants (from reference)
#define BB   256
#define CC   112
#define PP   200
#define DD   10
#define NCLS 200
#define MAXD 256.0f
#define EPSV 1e-8f
#define LSTR 202   // padded LDS row stride (dwords): even (b64 align), 2-way bank conflict max

typedef float v2f __attribute__((ext_vector_type(2)));
typedef float v8f __attribute__((ext_vector_type(8)));

// ---------------- Kernel 1: rowsum[r][p] = sum_d pp[r,p,d]; n[r][i] = ||pp[r,:,i]|| ----
__global__ __launch_bounds__(256) void k_rowsum(const float* __restrict__ pp,
                                                float* __restrict__ rowsum,
                                                float* __restrict__ nmat) {
    const int r = blockIdx.x;          // 0..223
    const int t = threadIdx.x;
    const float* base = pp + (size_t)r * (PP * DD);
    if (t < PP) {
        float s = 0.0f;
        #pragma unroll
        for (int d = 0; d < DD; ++d) s += base[t * DD + d];
        rowsum[r * PP + t] = s;
    } else if (t < PP + 16) {
        const int i = t - PP;
        float s = 0.0f;
        if (i < DD) {
            for (int p = 0; p < PP; ++p) { float v = base[p * DD + i]; s += v * v; }
        }
        nmat[r * 16 + i] = sqrtf(s);
    }
}

// ---------------- Kernel 2: exact lax.top_k membership masks per row ------------------
// clst mask: top-10 of rowsum ; sep mask: top-190 of (D - rowsum) (== bottom-190 of rowsum)
// tie-break identical to lax.top_k: equal values -> lower index first.
__global__ __launch_bounds__(256) void k_masks(const float* __restrict__ rowsum,
                                               float* __restrict__ mclst,
                                               float* __restrict__ msep) {
    const int r = blockIdx.x;
    const int t = threadIdx.x;
    __shared__ float rs[PP];
    if (t < PP) rs[t] = rowsum[r * PP + t];
    __syncthreads();
    if (t < PP) {
        const float v = rs[t];
        int rank_hi = 0, rank_lo = 0;
        for (int q = 0; q < PP; ++q) {
            const float vq = rs[q];
            const int eqlt = (vq == v) && (q < t);
            rank_hi += ((vq > v) || eqlt) ? 1 : 0;   // rank among descending rowsum
            rank_lo += ((vq < v) || eqlt) ? 1 : 0;   // rank among descending (D - rowsum)
        }
        mclst[r * PP + t] = (rank_hi < DD) ? 1.0f : 0.0f;
        msep [r * PP + t] = (rank_lo < (PP - DD)) ? 1.0f : 0.0f;
    }
}

// ---------------- Kernel 3: WMMA Gram matrices for ortho_p / ortho_c ------------------
// bid < 224 : self-Gram   G[c]  = pp[c]^T pp[c]          (ortho_p partial)
// bid >= 224: cross-Gram  Gc[c][i][j] = sum_p neg[c,p,i]*pos[c,p,j]  (ortho_c partial)
// One wave32 per matrix; operand tiles staged to LDS transposed+zero-padded to 16 rows,
// so the K-loop is 50x { 2x ds_load_b64 ; v_wmma_f32_16x16x4_f32 } with full EXEC,
// no per-iteration global waits, no EXEC-mask branches.
__global__ __launch_bounds__(32) void k_ortho(const float* __restrict__ pp,
                                              const float* __restrict__ nmat,
                                              float* __restrict__ partials) {
    __shared__ __align__(16) float ldsA[16 * LSTR];
    __shared__ __align__(16) float ldsB[16 * LSTR];

    const int bid = blockIdx.x;                  // 0..335
    const bool cross = (bid >= 2 * CC);
    int cA, cB;
    if (!cross) { cA = bid; cB = bid; }                       // self
    else { const int c = bid - 2 * CC; cA = c + CC; cB = c; } // A = neg (i), B = pos (j)
    const float* Abase = pp + (size_t)cA * (PP * DD);    // [p][d]
    const float* Bbase = pp + (size_t)cB * (PP * DD);

    const int lane = threadIdx.x;
    const int half = lane >> 4;      // 0 or 1
    const int mi   = lane & 15;      // M for A-operand, N for B-operand

    // Zero the pad rows (d = 10..15) once.
    for (int i = 10 * LSTR + lane; i < 16 * LSTR; i += 32) {
        ldsA[i] = 0.0f;
        ldsB[i] = 0.0f;
    }
    // Stage transposed: lds[d * LSTR + p] = src[p * DD + d]  (coalesced global reads)
    for (int i = lane; i < PP * DD; i += 32) {
        const int p = i / DD, d = i % DD;
        ldsA[d * LSTR + p] = Abase[i];
    }
    if (cross) {
        for (int i = lane; i < PP * DD; i += 32) {
            const int p = i / DD, d = i % DD;
            ldsB[d * LSTR + p] = Bbase[i];
        }
    }
    __syncthreads();
    const float* LB = cross ? ldsB : ldsA;

    // Per-lane operand row base: A[m,k] lives at ldsA[mi*LSTR + k]; k pairs are contiguous.
    const float* rowA = ldsA + mi * LSTR;
    const float* rowB = LB   + mi * LSTR;

    v8f acc = {};
    for (int kb = 0; kb < PP; kb += 4) {
        const int k0 = kb + half * 2;                 // even -> 8B-aligned ds_load_b64
        const v2f a = *(const v2f*)(rowA + k0);
        const v2f b = *(const v2f*)(rowB + k0);
        acc = __builtin_amdgcn_wmma_f32_16x16x4_f32(false, a, false, b,
                                                    (short)0, acc, false, false);
    }

    // C/D layout: lane holds column N = mi; VGPR v holds row M = v + 8*half
    const float* nA = nmat + cA * 16;   // row norms (i)
    const float* nB = nmat + cB * 16;   // col norms (j)
    float s = 0.0f;
    if (mi < DD) {
        const float nj = nB[mi];
        #pragma unroll
        for (int v = 0; v < 8; ++v) {
            const int m = v + 8 * half;
            if (m < DD) s += acc[v] / fmaxf(nA[m] * nj, EPSV);
        }
    }
    __shared__ float red[32];
    red[lane] = s;
    __syncthreads();
    for (int off = 16; off > 0; off >>= 1) {
        if (lane < off) red[lane] += red[lane + off];
        __syncthreads();
    }
    if (lane == 0) partials[bid] = red[0];
}

// ---------------- Kernel 4: BCE column means (writes d_out[1..112]) -------------------
__global__ __launch_bounds__(256) void k_bce(const float* __restrict__ x,
                                             const float* __restrict__ y,
                                             float* __restrict__ out) {
    const int c = blockIdx.x;    // 0..111
    const int b = threadIdx.x;   // 0..255 == B
    const float xv = x[b * CC + c];
    const float yv = y[b * CC + c];
    const float sp = fmaxf(xv, 0.0f) + log1pf(expf(-fabsf(xv)));  // stable softplus
    __shared__ float red[256];
    red[b] = sp - xv * yv;
    __syncthreads();
    for (int off = 128; off > 0; off >>= 1) {
        if (b < off) red[b] += red[b + off];
        __syncthreads();
    }
    if (b == 0) out[1 + c] = red[0] / (float)BB;
}

// ---------------- Kernel 5: clst/sep masked-max pooling per batch element -------------
__global__ __launch_bounds__(128) void k_pool(const float* __restrict__ ctrue,
                                              const float* __restrict__ mind,
                                              const float* __restrict__ mclst,
                                              const float* __restrict__ msep,
                                              float* __restrict__ clst_b,
                                              float* __restrict__ sep_b) {
    const int b = blockIdx.x;
    const int t = threadIdx.x;
    __shared__ float mdinv[PP];
    for (int p = t; p < PP; p += 128) mdinv[p] = MAXD - mind[b * PP + p];
    __syncthreads();
    float mc = 0.0f, ms = 0.0f;
    if (t < CC) {
        const int c = t;
        const bool tr = ctrue[b * CC + c] > 0.5f;
        const int rc = tr ? c : (c + CC);        // idx     row for clst
        const int rs = tr ? (c + CC) : c;        // inv_idx row for sep
        const float* maC = mclst + rc * PP;
        const float* maS = msep  + rs * PP;
        for (int p = 0; p < PP; ++p) {
            mc = fmaxf(mc, mdinv[p] * maC[p]);
            ms = fmaxf(ms, mdinv[p] * maS[p]);
        }
    }
    __shared__ float r1[128];
    __shared__ float r2[128];
    r1[t] = mc; r2[t] = ms;
    __syncthreads();
    for (int off = 64; off > 0; off >>= 1) {
        if (t < off) { r1[t] += r1[t + off]; r2[t] += r2[t + off]; }
        __syncthreads();
    }
    if (t == 0) {
        clst_b[b] = MAXD - r1[0] / (float)CC;
        sep_b[b]  = MAXD - r2[0] / (float)CC;
    }
}

// ---------------- Kernel 6: per-row -log_softmax at target class ----------------------
__global__ __launch_bounds__(256) void k_target(const float* __restrict__ logits,
                                                const int* __restrict__ tt,
                                                float* __restrict__ tgt_b) {
    const int b = blockIdx.x;
    const int t = threadIdx.x;
    const float v = (t < NCLS) ? logits[b * NCLS + t] : -3.402823466e38f;
    __shared__ float red[256];
    red[t] = v;
    __syncthreads();
    for (int off = 128; off > 0; off >>= 1) {
        if (t < off) red[t] = fmaxf(red[t], red[t + off]);
        __syncthreads();
    }
    const float mx = red[0];
    __syncthreads();
    red[t] = (t < NCLS) ? expf(v - mx) : 0.0f;
    __syncthreads();
    for (int off = 128; off > 0; off >>= 1) {
        if (t < off) red[t] += red[t + off];
        __syncthreads();
    }
    if (t == 0) {
        const float lse = logf(red[0]) + mx;
        tgt_b[b] = -(logits[b * NCLS + tt[b]] - lse);
    }
}

// ---------------- Kernel 7: L1 (fixed-order, single block) ----------------------------
__global__ __launch_bounds__(256) void k_l1(const float* __restrict__ w,
                                            const float* __restrict__ m,
                                            float* __restrict__ l1out) {
    const int t = threadIdx.x;
    float s = 0.0f;
    for (int i = t; i < 2 * CC * PP; i += 256) s += fabsf(w[i] * m[i]);
    __shared__ float red[256];
    red[t] = s;
    __syncthreads();
    for (int off = 128; off > 0; off >>= 1) {
        if (t < off) red[t] += red[t + off];
        __syncthreads();
    }
    if (t == 0) *l1out = red[0];
}

// ---------------- Kernel 8: finalize — combine all scalars, fill d_out ----------------
__global__ __launch_bounds__(256) void k_final(const float* __restrict__ ortho_part, // 336
                                               const float* __restrict__ clst_b,     // 256
                                               const float* __restrict__ sep_b,      // 256
                                               const float* __restrict__ tgt_b,      // 256
                                               const float* __restrict__ l1p,        // 1
                                               float* __restrict__ out) {
    const int t = threadIdx.x;
    __shared__ float red[256];
    float sum_p, sum_c, sum_clst, sum_sep, sum_tgt, sum_e;

    red[t] = (t < 2 * CC) ? ortho_part[t] : 0.0f;
    __syncthreads();
    for (int off = 128; off > 0; off >>= 1) { if (t < off) red[t] += red[t + off]; __syncthreads(); }
    sum_p = red[0]; __syncthreads();

    red[t] = (t < CC) ? ortho_part[2 * CC + t] : 0.0f;
    __syncthreads();
    for (int off = 128; off > 0; off >>= 1) { if (t < off) red[t] += red[t + off]; __syncthreads(); }
    sum_c = red[0]; __syncthreads();

    red[t] = clst_b[t];
    __syncthreads();
    for (int off = 128; off > 0; off >>= 1) { if (t < off) red[t] += red[t + off]; __syncthreads(); }
    sum_clst = red[0]; __syncthreads();

    red[t] = sep_b[t];
    __syncthreads();
    for (int off = 128; off > 0; off >>= 1) { if (t < off) red[t] += red[t + off]; __syncthreads(); }
    sum_sep = red[0]; __syncthreads();

    red[t] = tgt_b[t];
    __syncthreads();
    for (int off = 128; off > 0; off >>= 1) { if (t < off) red[t] += red[t + off]; __syncthreads(); }
    sum_tgt = red[0]; __syncthreads();

    red[t] = (t < CC) ? out[1 + t] : 0.0f;   // concepts_loss already in d_out
    __syncthreads();
    for (int off = 128; off > 0; off >>= 1) { if (t < off) red[t] += red[t + off]; __syncthreads(); }
    sum_e = red[0];

    if (t == 0) {
        const float ortho_p = sum_p / (float)(DD * CC * 2) - 1.0f;
        const float ortho_c = sum_c / (float)(DD * CC);
        const float clst    = sum_clst / (float)BB;
        const float sep     = sum_sep  / (float)BB;
        const float tgt     = sum_tgt  / (float)BB;
        const float entropy = sum_e    / (float)CC;
        const float l1      = *l1p;
        const float summed  = 1.0f * entropy + 0.8f * clst + (-0.08f) * sep
                            + 1e-4f * l1 + 1.0f * ortho_p + 1.0f * ortho_c;
        out[0]   = tgt;            // target_loss
        // out[1..112] = concepts_loss (written by k_bce)
        out[113] = summed;
        out[114] = tgt + summed;   // total
        out[115] = entropy;
        out[116] = clst;
        out[117] = sep;
        out[118] = l1;
        out[119] = ortho_p;
        out[120] = ortho_c;
    }
}

extern "C" void kernel_launch(void* const* d_in, const int* in_sizes, int n_in,
                              void* d_out, int out_size, void* d_ws, size_t ws_size,
                              hipStream_t stream) {
    const float* concepts_pred = (const float*)d_in[0];   // (B,C)
    const float* concepts_true = (const float*)d_in[1];   // (B,C)
    const float* min_dist      = (const float*)d_in[2];   // (B,P)
    const float* proto         = (const float*)d_in[3];   // (2C,P,D)
    const float* tgt_logits    = (const float*)d_in[4];   // (B,NCLS)
    const int*   tgt_true      = (const int*)d_in[5];     // (B,)
    const float* llw           = (const float*)d_in[6];   // (2C,P)
    const float* l1m           = (const float*)d_in[7];   // (2C,P)
    float* out = (float*)d_out;

    // Workspace layout (floats)
    float* ws        = (float*)d_ws;
    float* rowsum    = ws;                        // 224*200 = 44800
    float* nmat      = rowsum + 2 * CC * PP;      // 224*16  =  3584
    float* mclst     = nmat + 2 * CC * 16;        // 44800
    float* msep      = mclst + 2 * CC * PP;       // 44800
    float* opart     = msep + 2 * CC * PP;        // 336
    float* clst_b    = opart + 3 * CC;            // 256
    float* sep_b     = clst_b + BB;               // 256
    float* tgt_b     = sep_b + BB;                // 256
    float* l1s       = tgt_b + BB;                // 1

    k_rowsum<<<2 * CC, 256, 0, stream>>>(proto, rowsum, nmat);
    k_masks <<<2 * CC, 256, 0, stream>>>(rowsum, mclst, msep);
    k_ortho <<<3 * CC, 32, 0, stream>>>(proto, nmat, opart);
    k_bce   <<<CC, 256, 0, stream>>>(concepts_pred, concepts_true, out);
    k_pool  <<<BB, 128, 0, stream>>>(concepts_true, min_dist, mclst, msep, clst_b, sep_b);
    k_target<<<BB, 256, 0, stream>>>(tgt_logits, tgt_true, tgt_b);
    k_l1    <<<1, 256, 0, stream>>>(llw, l1m, l1s);
    k_final <<<1, 256, 0, stream>>>(opart, clst_b, sep_b, tgt_b, l1s, out);
}